// DRewGatedGCNGraphGymLayer_48739288875406
// MI455X (gfx1250) — compile-verified
//
#include <hip/hip_runtime.h>
#include <hip/hip_bf16.h>

typedef __attribute__((ext_vector_type(2))) float v2f;
typedef __attribute__((ext_vector_type(8))) float v8f;

#define N_NODES 50000
#define N_EDGES 800000
#define DIM     64
#define KHOP    3
#define AGG_EPS 1e-6f
#define BN_EPS  1e-5f

// ---------------------------------------------------------------------------
// Kernel 1: five node-level GEMMs via V_WMMA_F32_16X16X4_F32.
//   which==0: Ax = xs[2] @ Wa + ba   -> written straight into d_out
//   which==1: Dx = xs[2] @ Wd + bd
//   which==2..4: Bx[k] = xs[2-k] @ Wb[k] + bb[k]   (state_idx = [2,1,0])
// One wave (32 lanes) computes one 16x64 output tile.
// ---------------------------------------------------------------------------
__global__ __launch_bounds__(32) void node_gemm_wmma(
    const float* __restrict__ xs,                      // [3, N, 64]
    const float* __restrict__ Wa, const float* __restrict__ ba,
    const float* __restrict__ Wd, const float* __restrict__ bd,
    const float* __restrict__ Wb, const float* __restrict__ bb,
    float* __restrict__ Ax, float* __restrict__ Dx, float* __restrict__ Bx)
{
    const int lane = threadIdx.x;      // 0..31
    const int half = lane >> 4;        // 0 or 1
    const int lm   = lane & 15;        // 0..15
    const int m0   = blockIdx.x * 16;  // node tile (N divisible by 16)
    const int which = blockIdx.y;      // 0..4

    const float* xin;
    const float* W;
    const float* bias;
    float* outp;
    if (which == 0)      { xin = xs + (size_t)2 * N_NODES * DIM; W = Wa; bias = ba; outp = Ax; }
    else if (which == 1) { xin = xs + (size_t)2 * N_NODES * DIM; W = Wd; bias = bd; outp = Dx; }
    else {
        const int k = which - 2;                       // hop 0..2
        xin  = xs + (size_t)(2 - k) * N_NODES * DIM;   // delayed state
        W    = Wb + (size_t)k * DIM * DIM;
        bias = bb + k * DIM;
        outp = Bx + (size_t)k * N_NODES * DIM;
    }

    // Load all 16 A-fragments (16x4 f32 tiles) for this node tile once.
    // Lane layout per ISA: lanes 0-15 hold K={4k+0,4k+1}, lanes 16-31 K={4k+2,4k+3}.
    v2f afrag[16];
    const float* arow = xin + (size_t)(m0 + lm) * DIM + 2 * half;
    #pragma unroll
    for (int kk = 0; kk < 16; ++kk)
        afrag[kk] = *(const v2f*)(arow + kk * 4);       // 8-byte aligned

    #pragma unroll
    for (int n0 = 0; n0 < DIM; n0 += 16) {
        v8f c = {};
        #pragma unroll
        for (int kk = 0; kk < 16; ++kk) {
            // B fragment: rows K={4kk+2*half, +1} of W, cols n0+lm (stride DIM)
            const float* wp = W + (size_t)(kk * 4 + 2 * half) * DIM + n0 + lm;
            v2f b;
            b.x = wp[0];
            b.y = wp[DIM];
            c = __builtin_amdgcn_wmma_f32_16x16x4_f32(
                    /*neg_a=*/false, afrag[kk],
                    /*neg_b=*/false, b,
                    /*c_mod=*/(short)0, c,
                    /*reuse_a=*/false, /*reuse_b=*/false);
        }
        const float bv = bias[n0 + lm];
        #pragma unroll
        for (int r = 0; r < 8; ++r) {
            const int m = m0 + r + 8 * half;            // C layout: M = r + 8*half
            outp[(size_t)m * DIM + n0 + lm] = c[r] + bv;
        }
    }
}

// ---------------------------------------------------------------------------
// Kernel 2: edge stage. One wave per edge; each lane handles 2 dims (float2).
// sigma = sigmoid(Dx[i] + Bx[k][j]); scatter num += sigma*Bx, den += sigma
// into per-(node,hop) slots via f32 atomics (L2-resident: num+den = 76.8 MB).
// ---------------------------------------------------------------------------
__global__ __launch_bounds__(256) void edge_gate_scatter(
    const int* __restrict__ edge_index,   // [2, E] (row0 = src j, row1 = dst i)
    const int* __restrict__ edge_attr,    // [E] hop in 1..3
    const float* __restrict__ Dx,         // [N, 64]
    const float* __restrict__ Bx,         // [3, N, 64]
    float* __restrict__ num, float* __restrict__ den)   // [N, 3, 64]
{
    const int t    = blockIdx.x * 256 + threadIdx.x;
    const int e    = t >> 5;
    const int lane = t & 31;
    if (e >= N_EDGES) return;

    const int j = edge_index[e];
    const int i = edge_index[N_EDGES + e];
    const int k = edge_attr[e] - 1;
    const int d = lane * 2;

    const v2f dxi = *(const v2f*)(Dx + (size_t)i * DIM + d);
    const v2f bxj = *(const v2f*)(Bx + ((size_t)k * N_NODES + j) * DIM + d);

    const float sx = dxi.x + bxj.x;
    const float sy = dxi.y + bxj.y;
    const float gx = 1.0f / (1.0f + __expf(-sx));
    const float gy = 1.0f / (1.0f + __expf(-sy));

    float* np = num + ((size_t)i * KHOP + k) * DIM + d;
    float* dp = den + ((size_t)i * KHOP + k) * DIM + d;
    atomicAdd(np + 0, gx * bxj.x);
    atomicAdd(np + 1, gy * bxj.y);
    atomicAdd(dp + 0, gx);
    atomicAdd(dp + 1, gy);
}

// ---------------------------------------------------------------------------
// Kernel 3: eta = num/(den+eps); x_new = Ax + (1/3)*sum_k eta  (in d_out,
// which already holds Ax). Accumulate per-feature sum / sum-of-squares via
// LDS atomics, then one global atomic per (block, feature).
// Block = 256 threads = 4 nodes x 64 dims.
// ---------------------------------------------------------------------------
__global__ __launch_bounds__(256) void combine_stats(
    const float* __restrict__ num, const float* __restrict__ den,
    float* __restrict__ xnew,                       // d_out (holds Ax on entry)
    float* __restrict__ gsum, float* __restrict__ gsumsq)
{
    __shared__ float s_sum[DIM];
    __shared__ float s_sq[DIM];
    const int tx  = threadIdx.x;
    const int d   = tx & 63;
    const int row = tx >> 6;
    if (tx < DIM) { s_sum[tx] = 0.0f; s_sq[tx] = 0.0f; }
    __syncthreads();

    const int n = blockIdx.x * 4 + row;             // N divisible by 4
    float eta = 0.0f;
    #pragma unroll
    for (int k = 0; k < KHOP; ++k) {
        const size_t off = ((size_t)n * KHOP + k) * DIM + d;
        eta += num[off] / (den[off] + AGG_EPS);
    }
    const size_t o = (size_t)n * DIM + d;
    const float v = xnew[o] + (1.0f / 3.0f) * eta;
    xnew[o] = v;

    atomicAdd(&s_sum[d], v);
    atomicAdd(&s_sq[d], v * v);
    __syncthreads();
    if (tx < DIM) {
        atomicAdd(&gsum[tx],   s_sum[tx]);
        atomicAdd(&gsumsq[tx], s_sq[tx]);
    }
}

// ---------------------------------------------------------------------------
// Kernel 4: BatchNorm (training-mode batch stats, biased var) + ReLU in place.
// ---------------------------------------------------------------------------
__global__ __launch_bounds__(256) void bn_relu(
    float* __restrict__ y,
    const float* __restrict__ gsum, const float* __restrict__ gsumsq,
    const float* __restrict__ gamma, const float* __restrict__ beta)
{
    const int t = blockIdx.x * 256 + threadIdx.x;
    if (t >= N_NODES * DIM) return;
    const int d = t & 63;
    const float inv_n = 1.0f / (float)N_NODES;
    const float mean  = gsum[d] * inv_n;
    const float var   = gsumsq[d] * inv_n - mean * mean;
    float v = (y[t] - mean) * rsqrtf(var + BN_EPS);
    v = gamma[d] * v + beta[d];
    y[t] = fmaxf(v, 0.0f);
}

// ---------------------------------------------------------------------------
extern "C" void kernel_launch(void* const* d_in, const int* in_sizes, int n_in,
                              void* d_out, int out_size, void* d_ws, size_t ws_size,
                              hipStream_t stream)
{
    const float* xs         = (const float*)d_in[0];
    const int*   edge_index = (const int*)  d_in[1];
    const int*   edge_attr  = (const int*)  d_in[2];
    const float* Wa         = (const float*)d_in[3];
    const float* ba         = (const float*)d_in[4];
    const float* Wd         = (const float*)d_in[5];
    const float* bd         = (const float*)d_in[6];
    const float* Wb         = (const float*)d_in[7];
    const float* bb         = (const float*)d_in[8];
    const float* gamma      = (const float*)d_in[9];
    const float* beta       = (const float*)d_in[10];

    float* out = (float*)d_out;
    float* ws  = (float*)d_ws;

    const size_t ND = (size_t)N_NODES * DIM;
    float* Dx     = ws;                 // N*64
    float* Bx     = Dx  + ND;           // 3*N*64
    float* num    = Bx  + 3 * ND;       // 3*N*64
    float* den    = num + 3 * ND;       // 3*N*64
    float* gsum   = den + 3 * ND;       // 64
    float* gsumsq = gsum + DIM;         // 64

    // Zero the accumulator region (num | den | gsum | gsumsq) — capture-safe.
    hipMemsetAsync(num, 0, (6 * ND + 2 * DIM) * sizeof(float), stream);

    // 1) Five GEMMs (Ax -> d_out, Dx, Bx[0..2]) with f32 WMMA.
    dim3 g1(N_NODES / 16, 5);
    node_gemm_wmma<<<g1, 32, 0, stream>>>(xs, Wa, ba, Wd, bd, Wb, bb, out, Dx, Bx);

    // 2) Edge gating + scatter (wave-per-edge).
    const int edge_threads = N_EDGES * 32;
    edge_gate_scatter<<<edge_threads / 256, 256, 0, stream>>>(
        edge_index, edge_attr, Dx, Bx, num, den);

    // 3) Combine + batch statistics.
    combine_stats<<<N_NODES / 4, 256, 0, stream>>>(num, den, out, gsum, gsumsq);

    // 4) BatchNorm + ReLU.
    bn_relu<<<(N_NODES * DIM + 255) / 256, 256, 0, stream>>>(out, gsum, gsumsq, gamma, beta);
}